// TopKActivation_90314572300677
// MI455X (gfx1250) — compile-verified
//
#include <hip/hip_runtime.h>
#include <stdint.h>

// Row-wise top-k(ReLU) masking for x:[16384,4096] f32, k=64.
// Memory-bound: 512 MB traffic -> ~22us floor @ 23.3 TB/s. No matrix math ->
// WMMA not applicable; use gfx1250 async global->LDS staging + LDS radix select.

#define THREADS 256
#define ROWLEN  4096
#define PER_THREAD (ROWLEN / THREADS)   // 16
#define VEC_PER_THREAD (ROWLEN / 4 / THREADS) // 4 x float4

typedef int v4i __attribute__((vector_size(4 * sizeof(int))));

__device__ __forceinline__ void async_copy_b128_to_lds(const float* gsrc, float* ldst) {
#if __has_builtin(__builtin_amdgcn_global_load_async_to_lds_b128)
    __builtin_amdgcn_global_load_async_to_lds_b128(
        (__attribute__((address_space(1))) v4i*)gsrc,
        (__attribute__((address_space(3))) v4i*)ldst,
        /*imm offset*/0, /*cpol*/0);
#else
    unsigned loff = (unsigned)(unsigned long long)(__attribute__((address_space(3))) char*)ldst;
    unsigned long long gaddr = (unsigned long long)gsrc;
    asm volatile("global_load_async_to_lds_b128 %0, %1, off"
                 :: "v"(loff), "v"(gaddr) : "memory");
#endif
}

__device__ __forceinline__ void wait_async_zero() {
#if __has_builtin(__builtin_amdgcn_s_wait_asynccnt)
    __builtin_amdgcn_s_wait_asynccnt(0);
#else
    asm volatile("s_wait_asynccnt 0" ::: "memory");
#endif
}

__global__ __launch_bounds__(THREADS)
void TopKActivation_kernel(const float* __restrict__ x,
                           const int* __restrict__ kptr,
                           float* __restrict__ out) {
    __shared__ __align__(16) float row[ROWLEN];      // 16 KB staged row
    __shared__ unsigned hist[256];
    __shared__ unsigned scan[256];
    __shared__ unsigned bc_bin;
    __shared__ unsigned bc_gt;

    const int tid = threadIdx.x;
    const long long rbase = (long long)blockIdx.x * (long long)ROWLEN;
    const float* grow = x + rbase;

    // --- Stage row into LDS with gfx1250 async loads (ASYNCcnt path) ---
#pragma unroll
    for (int j = 0; j < VEC_PER_THREAD; ++j) {
        const int v = tid + j * THREADS;             // float4 index
        async_copy_b128_to_lds(grow + v * 4, &row[v * 4]);
    }
    wait_async_zero();      // this wave's async loads done
    __syncthreads();        // all waves' loads visible

    int k = *kptr;
    if (k > ROWLEN) k = ROWLEN;
    if (k < 1) k = 1;
    unsigned remaining = (unsigned)k;
    unsigned prefix = 0;    // selected high bits of threshold so far

    // --- 4 x 8-bit radix select on relu-bits (monotone u32 for x>=0) ---
#pragma unroll
    for (int pass = 0; pass < 4; ++pass) {
        const int shift = 24 - pass * 8;

        hist[tid] = 0;
        __syncthreads();

#pragma unroll
        for (int j = 0; j < PER_THREAD; ++j) {
            const int i = tid + j * THREADS;         // lane-consecutive: no LDS bank conflicts
            const float f = row[i];
            const unsigned u = (f > 0.0f) ? __float_as_uint(f) : 0u;
            const bool match = (pass == 0) || ((u >> (shift + 8)) == prefix);
            if (match) atomicAdd(&hist[(u >> shift) & 255u], 1u);
        }
        __syncthreads();

        // Suffix scan over 256 bins: scan[t] = count of values in bins >= t
        scan[tid] = hist[tid];
        __syncthreads();
#pragma unroll
        for (int d = 1; d < 256; d <<= 1) {
            const unsigned v = scan[tid] + ((tid + d < 256) ? scan[tid + d] : 0u);
            __syncthreads();
            scan[tid] = v;
            __syncthreads();
        }

        const unsigned ge = scan[tid];
        const unsigned gt = (tid < 255) ? scan[tid + 1] : 0u;
        if (ge >= remaining && gt < remaining) {     // unique crossing bin
            bc_bin = (unsigned)tid;
            bc_gt  = gt;
        }
        __syncthreads();
        prefix = (prefix << 8) | bc_bin;
        remaining -= bc_gt;
        __syncthreads();
    }

    const unsigned T = prefix;                       // exact k-th largest relu bits

    // --- Output: keep relu(x) where relu_bits >= T (zeros stay zero) ---
    float* gout = out + rbase;
#pragma unroll
    for (int j = 0; j < PER_THREAD; ++j) {
        const int i = tid + j * THREADS;
        const float f = row[i];
        const unsigned u = (f > 0.0f) ? __float_as_uint(f) : 0u;
        gout[i] = (u >= T) ? __uint_as_float(u) : 0.0f;
    }
}

extern "C" void kernel_launch(void* const* d_in, const int* in_sizes, int n_in,
                              void* d_out, int out_size, void* d_ws, size_t ws_size,
                              hipStream_t stream) {
    const float* x = (const float*)d_in[0];
    const int*   k = (const int*)d_in[1];
    float* out = (float*)d_out;

    const int rows = in_sizes[0] / ROWLEN;           // 16384
    hipLaunchKernelGGL(TopKActivation_kernel, dim3(rows), dim3(THREADS), 0, stream,
                       x, k, out);
}